// HeteroDotProductPredictor_62105227100322
// MI455X (gfx1250) — compile-verified
//
#include <hip/hip_runtime.h>
#include <hip/hip_bf16.h>

typedef float v2f __attribute__((ext_vector_type(2)));
typedef float v8f __attribute__((ext_vector_type(8)));

// Per-edge dot(h[src], h[dst]) via V_WMMA_F32_16X16X4_F32 diagonal trick.
// 16 edges per wave32. A = 16 hu-rows (16x4 f32), B = 16 hv-rows as 4x16,
// diag(A x B) accumulated over K=128 in 32 WMMA chunks of K=4.
__global__ __launch_bounds__(256) void u_dot_v_wmma_kernel(
    const float* __restrict__ h,
    const long long* __restrict__ src,
    const long long* __restrict__ dst,
    float* __restrict__ out,
    int E)
{
    const int lane = threadIdx.x & 31;
    const int waveInBlock = threadIdx.x >> 5;
    const long long wave = (long long)blockIdx.x * (blockDim.x >> 5) + waveInBlock;
    const long long ebase = wave * 16;
    if (ebase >= E) return;                 // wave-uniform: EXEC stays all-1s

    const int e_lane = lane & 15;           // which of the 16 edges this lane serves
    const int half   = lane >> 4;           // 0: K slots 0,1   1: K slots 2,3

    long long e = ebase + e_lane;
    if (e >= E) e = (long long)E - 1;       // data clamp only; no divergence

    const long long s = src[e];
    const long long d = dst[e];
    const float* __restrict__ pu = h + s * 128 + half * 4;
    const float* __restrict__ pv = h + d * 128 + half * 4;

    v8f acc = {0.f, 0.f, 0.f, 0.f, 0.f, 0.f, 0.f, 0.f};

#pragma unroll
    for (int i = 0; i < 16; ++i) {
        // 16B-aligned b128 loads: row stride 512B, offsets are multiples of 16B.
        const float4 u4 = *(const float4*)(pu + i * 8);
        const float4 v4 = *(const float4*)(pv + i * 8);

        v2f a0 = {u4.x, u4.y};
        v2f b0 = {v4.x, v4.y};
        v2f a1 = {u4.z, u4.w};
        v2f b1 = {v4.z, v4.w};

        // D = A*B + C, f32 16x16x4; accumulate into acc.
        acc = __builtin_amdgcn_wmma_f32_16x16x4_f32(
            false, a0, false, b0, (short)0, acc, false, false);
        acc = __builtin_amdgcn_wmma_f32_16x16x4_f32(
            false, a1, false, b1, (short)0, acc, false, false);
    }

    // Diagonal extraction:
    //   lanes 0..7  : D[r][r]     in acc[r],  r = lane
    //   lanes 24..31: D[r+8][r+8] in acc[r],  r = lane-24
    const int comp = (lane < 8) ? lane : (lane - 24);
    float v = acc[0];
    v = (comp == 1) ? acc[1] : v;
    v = (comp == 2) ? acc[2] : v;
    v = (comp == 3) ? acc[3] : v;
    v = (comp == 4) ? acc[4] : v;
    v = (comp == 5) ? acc[5] : v;
    v = (comp == 6) ? acc[6] : v;
    v = (comp == 7) ? acc[7] : v;

    long long eo = -1;
    if (lane < 8)        eo = ebase + lane;
    else if (lane >= 24) eo = ebase + (lane - 16);
    if (eo >= 0 && eo < E) out[eo] = v;
}

extern "C" void kernel_launch(void* const* d_in, const int* in_sizes, int n_in,
                              void* d_out, int out_size, void* d_ws, size_t ws_size,
                              hipStream_t stream) {
    const float*     h   = (const float*)d_in[0];
    const long long* src = (const long long*)d_in[1];
    const long long* dst = (const long long*)d_in[2];
    float*           out = (float*)d_out;

    const int E = in_sizes[1];              // 640000 edges
    // 16 edges per wave, 8 waves (256 threads) per block -> 128 edges per block.
    const int blocks = (E + 127) / 128;
    u_dot_v_wmma_kernel<<<blocks, 256, 0, stream>>>(h, src, dst, out, E);
}